// DocRelPrompt_81999515615940
// MI455X (gfx1250) — compile-verified
//
#include <hip/hip_runtime.h>
#include <math.h>

// Problem constants (from reference)
#define B_    64
#define L_    512
#define H_    768
#define HEAD_ 64
#define NP_   10
#define NOUT_ (NP_ + L_)   // 522 rows in output per batch

typedef float v2f __attribute__((ext_vector_type(2)));
typedef float v8f __attribute__((ext_vector_type(8)));

// ---------------------------------------------------------------------------
// Kernel 1: fused streaming pass over hidden_states_src (the 100MB tensor).
//   out[b, 10+l, h] = hidden[b,l,h] + (1-r)*lab[0,h] + r*lab[1,h]
//   hacc[b,h]      += sum over this block's l-chunk of hidden[b,l,h]
// Bandwidth-bound: 1 read + 1 write per element, fully coalesced over h.
// ---------------------------------------------------------------------------
__global__ void k_stream(const float* __restrict__ hid,
                         const float* __restrict__ rel,
                         const float* __restrict__ lab,
                         float* __restrict__ out,
                         float* __restrict__ hacc) {
    const int h  = blockIdx.x * blockDim.x + threadIdx.x;   // 0..767
    const int l0 = blockIdx.y * 32;                          // 16 chunks of 32
    const int b  = blockIdx.z;
    const float r    = rel[b];
    const float hrel = (1.0f - r) * lab[h] + r * lab[H_ + h];
    const float* src = hid + ((size_t)b * L_ + l0) * H_ + h;
    float*       dst = out + ((size_t)b * NOUT_ + NP_ + l0) * H_ + h;
    float s = 0.0f;
#pragma unroll 8
    for (int i = 0; i < 32; ++i) {
        float v = src[(size_t)i * H_];
        s += v;
        dst[(size_t)i * H_] = v + hrel;
    }
    atomicAdd(&hacc[b * H_ + h], s);
}

// ---------------------------------------------------------------------------
// Kernel 2: one 20-wave workgroup computes (with fp32 WMMA 16x16x4):
//   waves  0-15: Kbar = (hacc/512) @ kw^T + kb      (64x64, K=768) -> LDS
//   waves 16-19: Q    = prompts(pad 16) @ qw^T + qb (16x64, K=768) -> LDS
//   waves  0-3 : S    = Q @ Kbar^T (16x64, K=64);  gate = sigmoid(S/8)
// VGPR layouts per CDNA5 ISA 7.12.2:
//   A 16x4 : v0 = A[M=lane%16][K+ (lane<16?0:2)], v1 = same K+1
//   B 4x16 : v0 = B[K+(lane<16?0:2)][N=lane%16],  v1 = same K+1
//   C 16x16: c[r] = C[M = r + (lane<16?0:8)][N = lane%16]
// All fragment loads are even-offset -> 8B-aligned v2f (b64) loads; the
// Q-row zero padding is done with clamped addresses + a 0/1 mask so the
// WMMA loops contain no divergent control flow.
// ---------------------------------------------------------------------------
__global__ __launch_bounds__(640) void k_gate(
    const float* __restrict__ hacc,
    const float* __restrict__ prompts,
    const float* __restrict__ qw, const float* __restrict__ qb,
    const float* __restrict__ kw, const float* __restrict__ kb,
    float* __restrict__ gate) {
    __shared__ float kbar_s[64 * 64];   // [b][d]
    __shared__ float q_s[16 * 64];      // [n][d] (rows >= 10 are padding)

    const int tid  = threadIdx.x;
#if defined(__HIP_DEVICE_COMPILE__) && __has_builtin(__builtin_amdgcn_wmma_f32_16x16x4_f32)
    const int w    = tid >> 5;
    const int lane = tid & 31;
    const int lm   = lane & 15;
    const int kh   = (lane >> 4) << 1;          // 0 or 2
    const int moff = (lane < 16) ? 0 : 8;

    if (w < 16) {
        // Kbar tile (m0 = b-rows, n0 = d-cols)
        const int m0 = (w & 3) * 16;
        const int n0 = (w >> 2) * 16;
        const float inv_l = 1.0f / (float)L_;
        const float* arow = hacc + (m0 + lm) * H_ + kh;
        const float* brow = kw   + (n0 + lm) * H_ + kh;
        v8f acc = {};
#pragma unroll 4
        for (int k = 0; k < H_; k += 4) {
            v2f a  = *(const v2f*)(arow + k);
            v2f bb = *(const v2f*)(brow + k);
            a.x *= inv_l; a.y *= inv_l;
            acc = __builtin_amdgcn_wmma_f32_16x16x4_f32(
                false, a, false, bb, (short)0, acc, false, false);
        }
        const float bias = kb[n0 + lm];
#pragma unroll
        for (int r = 0; r < 8; ++r)
            kbar_s[(m0 + moff + r) * 64 + (n0 + lm)] = acc[r] + bias;
    } else {
        // Q tile (rows n, zero-padded to 16 via clamped address + 0/1 mask)
        const int n0   = (w - 16) * 16;
        const int arn  = (lm < NP_) ? lm : 0;       // safe row to read
        const float am = (lm < NP_) ? 1.0f : 0.0f;  // zero-mask for pad rows
        const float* arow = prompts + arn * H_ + kh;
        const float* brow = qw + (n0 + lm) * H_ + kh;
        v8f acc = {};
#pragma unroll 4
        for (int k = 0; k < H_; k += 4) {
            v2f a  = *(const v2f*)(arow + k);
            v2f bb = *(const v2f*)(brow + k);
            a.x *= am; a.y *= am;
            acc = __builtin_amdgcn_wmma_f32_16x16x4_f32(
                false, a, false, bb, (short)0, acc, false, false);
        }
        const float bias = qb[n0 + lm];
#pragma unroll
        for (int r = 0; r < 8; ++r)
            q_s[(moff + r) * 64 + (n0 + lm)] = acc[r] + bias;
    }
    __syncthreads();
    if (w < 4) {
        // S[n, b] for b in [b0, b0+16); A = Q (M=n, K=d), B[d][b] = Kbar[b][d]
        const int b0 = w * 16;
        const float* arow = &q_s[lm * 64 + kh];
        const float* brow = &kbar_s[(b0 + lm) * 64 + kh];
        v8f acc = {};
#pragma unroll
        for (int k = 0; k < 64; k += 4) {
            v2f a  = *(const v2f*)(arow + k);
            v2f bb = *(const v2f*)(brow + k);
            acc = __builtin_amdgcn_wmma_f32_16x16x4_f32(
                false, a, false, bb, (short)0, acc, false, false);
        }
#pragma unroll
        for (int r = 0; r < 8; ++r) {
            const int n = moff + r;
            if (n < NP_) {
                const float sc = acc[r] * 0.125f;   // 1/sqrt(64)
                gate[(b0 + lm) * NP_ + n] = 1.0f / (1.0f + expf(-sc));
            }
        }
    }
#else
    // Scalar fallback (also what the host pass parses): same math, 640 threads.
    const float inv_l = 1.0f / (float)L_;
    for (int i = tid; i < 64 * 64; i += 640) {
        const int bi = i >> 6, d = i & 63;
        float s = 0.0f;
        for (int h = 0; h < H_; ++h) s += hacc[bi * H_ + h] * kw[d * H_ + h];
        kbar_s[i] = s * inv_l + kb[d];
    }
    for (int i = tid; i < NP_ * 64; i += 640) {
        const int n = i >> 6, d = i & 63;
        float s = 0.0f;
        for (int h = 0; h < H_; ++h) s += prompts[n * H_ + h] * qw[d * H_ + h];
        q_s[i] = s + qb[d];
    }
    __syncthreads();
    for (int i = tid; i < B_ * NP_; i += 640) {
        const int bi = i / NP_, n = i % NP_;
        float s = 0.0f;
        for (int d = 0; d < 64; ++d) s += q_s[n * 64 + d] * kbar_s[bi * 64 + d];
        gate[i] = 1.0f / (1.0f + expf(-s * 0.125f));
    }
#endif
}

// ---------------------------------------------------------------------------
// Kernel 3: out[b, n, h] = prompts[n,h] * gate[b,n]   (2 MB, many blocks)
// ---------------------------------------------------------------------------
__global__ void k_docprompts(const float* __restrict__ prompts,
                             const float* __restrict__ gate,
                             float* __restrict__ out) {
    const int idx = blockIdx.x * blockDim.x + threadIdx.x;
    if (idx >= B_ * NP_ * H_) return;
    const int h = idx % H_;
    const int n = (idx / H_) % NP_;
    const int b = idx / (H_ * NP_);
    out[((size_t)b * NOUT_ + n) * H_ + h] = prompts[n * H_ + h] * gate[b * NP_ + n];
}

extern "C" void kernel_launch(void* const* d_in, const int* in_sizes, int n_in,
                              void* d_out, int out_size, void* d_ws, size_t ws_size,
                              hipStream_t stream) {
    (void)in_sizes; (void)n_in; (void)out_size; (void)ws_size;
    const float* rel     = (const float*)d_in[0];   // (B,)
    const float* hid     = (const float*)d_in[1];   // (B, L, H)
    const float* prompts = (const float*)d_in[2];   // (10, H)
    const float* lab     = (const float*)d_in[3];   // (2, H)
    const float* qw      = (const float*)d_in[4];   // ref_qw (64, H)
    const float* qb      = (const float*)d_in[5];   // ref_qb (64,)
    const float* kw      = (const float*)d_in[6];   // ref_kw (64, H)
    const float* kb      = (const float*)d_in[7];   // ref_kb (64,)
    // d_in[8..11] (lab_*) are dead in the reference output.

    float* out  = (float*)d_out;                    // (B, 522, H)
    float* hacc = (float*)d_ws;                     // B*H accumulators
    float* gate = hacc + (size_t)B_ * H_;           // B*NP gates

    // zero the column-sum accumulators (graph-capture-safe)
    hipMemsetAsync(d_ws, 0, (size_t)B_ * H_ * sizeof(float), stream);

    // Pass 1: fused elementwise + column-sum reduction (3*16*64 = 3072 blocks)
    dim3 g1(H_ / 256, L_ / 32, B_);
    k_stream<<<g1, 256, 0, stream>>>(hid, rel, lab, out, hacc);

    // Pass 2: WMMA micro-GEMMs + sigmoid gates (one 20-wave workgroup)
    k_gate<<<1, 640, 0, stream>>>(hacc, prompts, qw, qb, kw, kb, gate);

    // Pass 3: gated prompt broadcast
    const int tot = B_ * NP_ * H_;
    k_docprompts<<<(tot + 255) / 256, 256, 0, stream>>>(prompts, gate, out);
}